// QuantumLayer_82377472737507
// MI455X (gfx1250) — compile-verified
//
#include <hip/hip_runtime.h>

// ---------------------------------------------------------------------------
// 10-qubit batched state-vector sim, one batch element per wave32.
// State: 1024 complex amplitudes; k bits[4:0] = register index (qubits 9..5),
//        k bits[9:5] = lane index   (qubits 4..0).
// Input RX layer fused into weight-layer-0 (RX(a)RX(b) = RX(a+b)): 20 gates.
// State packed as v2f (adjacent amplitude pairs) -> V_PK_FMA_F32 gate math.
// Cross-lane exchanges via DPP8 / DPP16 row_ror:8 / v_permlanex16 (pure VALU);
// lane-controlled CNOTs run EXEC-predicated (permutation maps the active lane
// set onto itself, so masked DPP is safe and needs no cndmask).
// Final probs @ signs^T via V_WMMA_F32_16X16X4_F32 (M=8, N=10, K=1024).
// ---------------------------------------------------------------------------

typedef __attribute__((ext_vector_type(2))) float v2f;
typedef __attribute__((ext_vector_type(8))) float v8f;

#define DEVINL __device__ __forceinline__

DEVINL float bcast(float v, int srclane) {
  return __int_as_float(__builtin_amdgcn_readlane(__float_as_int(v), srclane));
}

// Lane-XOR exchange, pure VALU (DPP), no memory counters involved.
template <int XM>
DEVINL float lane_xor(float v) {
  static_assert(XM == 1 || XM == 2 || XM == 4 || XM == 8 || XM == 16, "");
  int x = __float_as_int(v);
  if constexpr (XM < 8) {
    // DPP8: full permute within groups of 8 lanes; sel[i] = i ^ XM
    constexpr int imm =
        ((0 ^ XM) << 0) | ((1 ^ XM) << 3) | ((2 ^ XM) << 6) |
        ((3 ^ XM) << 9) | ((4 ^ XM) << 12) | ((5 ^ XM) << 15) |
        ((6 ^ XM) << 18) | ((7 ^ XM) << 21);
    x = __builtin_amdgcn_mov_dpp8(x, imm);
  } else if constexpr (XM == 8) {
    // DPP16 row_ror:8 — rotate by 8 within a 16-lane row == XOR 8
    x = __builtin_amdgcn_mov_dpp(x, 0x128, 0xF, 0xF, true);
  } else {
    // XOR 16: swap the two 16-lane halves (identity nibble selects)
    x = __builtin_amdgcn_permlanex16(x, x, 0x76543210, 0xFEDCBA98, false,
                                     false);
  }
  return __int_as_float(x);
}

template <int XM>
DEVINL v2f lane_xor2(v2f v) {
  v2f r;
  r.x = lane_xor<XM>(v.x);
  r.y = lane_xor<XM>(v.y);
  return r;
}

DEVINL v2f vfma(v2f a, v2f b, v2f c) { return __builtin_elementwise_fma(a, b, c); }
DEVINL v2f sp(float x) { v2f r; r.x = x; r.y = x; return r; }

// ---- RX(theta): new = c*self + s*(partner.im, -partner.re) ----------------
// State: cr[16], ci[16]; element i = amplitudes (2i, 2i+1).
// reg bit 0 (qubit 9): partner is the other component of the same element.
DEVINL void rx_reg0(v2f (&cr)[16], v2f (&ci)[16], float c, float s) {
  const v2f cs = sp(c), ss = sp(s);
#pragma unroll
  for (int i = 0; i < 16; ++i) {
    const v2f tr = cr[i], ti = ci[i];
    const v2f trs = tr.yx, tis = ti.yx;  // component-swapped partner
    cr[i] = vfma(cs, tr, ss * tis);
    ci[i] = vfma(cs, ti, -(ss * trs));
  }
}

// reg bits 1..4 (qubits 8..5): partner element i ^ (1 << (P-1)), elementwise
template <int P>
DEVINL void rx_regP(v2f (&cr)[16], v2f (&ci)[16], float c, float s) {
  static_assert(P >= 1 && P <= 4, "");
  constexpr int me = 1 << (P - 1);
  const v2f cs = sp(c), ss = sp(s);
#pragma unroll
  for (int i = 0; i < 16; ++i) {
    if (i & me) continue;
    const int q = i | me;
    const v2f t0r = cr[i], t0i = ci[i], t1r = cr[q], t1i = ci[q];
    cr[i] = vfma(cs, t0r, ss * t1i);
    ci[i] = vfma(cs, t0i, -(ss * t1r));
    cr[q] = vfma(cs, t1r, ss * t0i);
    ci[q] = vfma(cs, t1i, -(ss * t0r));
  }
}

// lane bits 0..4 (qubits 4..0): partner across lanes via DPP
template <int LB>
DEVINL void rx_lane(v2f (&cr)[16], v2f (&ci)[16], float c, float s) {
  const v2f cs = sp(c), ss = sp(s);
#pragma unroll
  for (int i = 0; i < 16; ++i) {
    const v2f yr = lane_xor2<(1 << LB)>(cr[i]);
    const v2f yi = lane_xor2<(1 << LB)>(ci[i]);
    cr[i] = vfma(cs, cr[i], ss * yi);
    ci[i] = vfma(cs, ci[i], -(ss * yr));
  }
}

// ---- CNOT variants ---------------------------------------------------------
// control/target both in element bits (reg bits >=1): compile-time elem swap
template <int EC, int ET>
DEVINL void cnot_rr_e(v2f (&cr)[16], v2f (&ci)[16]) {
  constexpr int tm = 1 << ET;
#pragma unroll
  for (int i = 0; i < 16; ++i) {
    if (!((i >> EC) & 1) || ((i >> ET) & 1)) continue;
    const int q = i | tm;
    v2f t = cr[i]; cr[i] = cr[q]; cr[q] = t;
    t = ci[i]; ci[i] = ci[q]; ci[q] = t;
  }
}

// control = reg bit 1, target = reg bit 0: component swap on odd elements
DEVINL void cnot_r1_r0(v2f (&cr)[16], v2f (&ci)[16]) {
#pragma unroll
  for (int i = 1; i < 16; i += 2) {
    cr[i] = cr[i].yx;
    ci[i] = ci[i].yx;
  }
}

// control = reg bit 0, target = lane bit 4: DPP xor-16 on .y components only
DEVINL void cnot_r0_l4(v2f (&cr)[16], v2f (&ci)[16]) {
#pragma unroll
  for (int i = 0; i < 16; ++i) {
    cr[i].y = lane_xor<16>(cr[i].y);
    ci[i].y = lane_xor<16>(ci[i].y);
  }
}

// control = lane bit LC, target = element bit ET: EXEC-predicated elem swap
template <int LC, int ET>
DEVINL void cnot_lr(v2f (&cr)[16], v2f (&ci)[16], int lane) {
  constexpr int tm = 1 << ET;
  if ((lane >> LC) & 1) {
#pragma unroll
    for (int i = 0; i < 16; ++i) {
      if ((i >> ET) & 1) continue;
      const int q = i | tm;
      v2f t = cr[i]; cr[i] = cr[q]; cr[q] = t;
      t = ci[i]; ci[i] = ci[q]; ci[q] = t;
    }
  }
}

// control & target both lane bits: EXEC-predicated DPP exchange.
// Safe: the XOR-(1<<LT) permutation preserves lane bit LC, so every source
// lane of an active lane is itself active.
template <int LC, int LT>
DEVINL void cnot_ll(v2f (&cr)[16], v2f (&ci)[16], int lane) {
  if ((lane >> LC) & 1) {
#pragma unroll
    for (int i = 0; i < 16; ++i) {
      cr[i] = lane_xor2<(1 << LT)>(cr[i]);
      ci[i] = lane_xor2<(1 << LT)>(ci[i]);
    }
  }
}

// ---- one RX layer over all 10 qubits (gate g0+j uses lane g0+j's cos/sin) --
DEVINL void rx_all10(v2f (&cr)[16], v2f (&ci)[16], float ch, float sh,
                     int g0) {
  // qubit j acts on k-bit (9-j): j=0..4 -> lane bits 4..0, j=5..9 -> reg 4..0
  rx_lane<4>(cr, ci, bcast(ch, g0 + 0), bcast(sh, g0 + 0));
  rx_lane<3>(cr, ci, bcast(ch, g0 + 1), bcast(sh, g0 + 1));
  rx_lane<2>(cr, ci, bcast(ch, g0 + 2), bcast(sh, g0 + 2));
  rx_lane<1>(cr, ci, bcast(ch, g0 + 3), bcast(sh, g0 + 3));
  rx_lane<0>(cr, ci, bcast(ch, g0 + 4), bcast(sh, g0 + 4));
  rx_regP<4>(cr, ci, bcast(ch, g0 + 5), bcast(sh, g0 + 5));
  rx_regP<3>(cr, ci, bcast(ch, g0 + 6), bcast(sh, g0 + 6));
  rx_regP<2>(cr, ci, bcast(ch, g0 + 7), bcast(sh, g0 + 7));
  rx_regP<1>(cr, ci, bcast(ch, g0 + 8), bcast(sh, g0 + 8));
  rx_reg0(cr, ci, bcast(ch, g0 + 9), bcast(sh, g0 + 9));
}

// ---- CNOT ring: (j, (j+1)%10) for j=0..9, bit positions (9-j, 8-j) / (0,9) -
DEVINL void cnot_ring(v2f (&cr)[16], v2f (&ci)[16], int lane) {
  cnot_ll<4, 3>(cr, ci, lane);  // j=0: k-bits (9,8)
  cnot_ll<3, 2>(cr, ci, lane);  // j=1: k-bits (8,7)
  cnot_ll<2, 1>(cr, ci, lane);  // j=2: k-bits (7,6)
  cnot_ll<1, 0>(cr, ci, lane);  // j=3: k-bits (6,5)
  cnot_lr<0, 3>(cr, ci, lane);  // j=4: k-bits (5,4) -> lane bit 0, elem bit 3
  cnot_rr_e<3, 2>(cr, ci);      // j=5: k-bits (4,3) -> elem bits (3,2)
  cnot_rr_e<2, 1>(cr, ci);      // j=6: k-bits (3,2) -> elem bits (2,1)
  cnot_rr_e<1, 0>(cr, ci);      // j=7: k-bits (2,1) -> elem bits (1,0)
  cnot_r1_r0(cr, ci);           // j=8: k-bits (1,0) -> comp swap, odd elems
  cnot_r0_l4(cr, ci);           // j=9: k-bits (0,9) -> xor16 on .y comps
}

// ---- one K=4 WMMA step of the (8x1024)x(1024x10) sign-GEMM -----------------
// A layout (f32 16x4): lane L<16 holds A[M=L][K=4kc+0], A[M=L][K=4kc+1];
//                      lane L>=16 holds K=4kc+2, 4kc+3. B mirrors (4x16).
DEVINL v8f gemm_step(const float* __restrict__ base, int shift, int half,
                     int kc, v8f c) {
  v2f a;
  a.x = base[4 * kc];
  a.y = base[4 * kc + 1];
  const int k0 = 4 * kc + 2 * half;
  v2f bm;
  bm.x = 1.0f - 2.0f * (float)((k0 >> shift) & 1);
  bm.y = 1.0f - 2.0f * (float)(((k0 + 1) >> shift) & 1);
  return __builtin_amdgcn_wmma_f32_16x16x4_f32(false, a, false, bm, (short)0,
                                               c, false, false);
}

__global__ __launch_bounds__(256) void qsim_kernel(
    const float* __restrict__ inp, const float* __restrict__ wts,
    float* __restrict__ out) {
  __shared__ __align__(16) float P[8][1024];  // per-wave probs, 32 KB

  const int lane = threadIdx.x & 31;
  const int wave = threadIdx.x >> 5;
  const int b = blockIdx.x * 8 + wave;

  // Gate fusion: RX(inputs[b,j]) followed (no CNOT between) by
  // RX(weights[0,j]) == RX(inputs[b,j] + weights[0,j]).
  // lane g in [0,10): fused layer-0 angle; lane g in [10,20): layer-1 angle.
  float ang = 0.0f;
  if (lane < 10)
    ang = inp[b * 10 + lane] + wts[lane];
  else if (lane < 20)
    ang = wts[lane];
  const float ch = __cosf(0.5f * ang);
  const float sh = __sinf(0.5f * ang);

  // |0...0>
  v2f cr[16], ci[16];
#pragma unroll
  for (int i = 0; i < 16; ++i) { cr[i] = sp(0.0f); ci[i] = sp(0.0f); }
  cr[0].x = (lane == 0) ? 1.0f : 0.0f;

  // circuit: [fused RX layer, ring] x then [layer-1 RX, ring]
  rx_all10(cr, ci, ch, sh, 0);
  cnot_ring(cr, ci, lane);
  rx_all10(cr, ci, ch, sh, 10);
  cnot_ring(cr, ci, lane);

  // probabilities -> LDS (k = lane*32 + 2i), 16B vector stores
#pragma unroll
  for (int j = 0; j < 8; ++j) {
    const v2f pa = vfma(cr[2 * j], cr[2 * j], ci[2 * j] * ci[2 * j]);
    const v2f pb =
        vfma(cr[2 * j + 1], cr[2 * j + 1], ci[2 * j + 1] * ci[2 * j + 1]);
    float4 v;
    v.x = pa.x; v.y = pa.y; v.z = pb.x; v.w = pb.y;
    *(float4*)&P[wave][lane * 32 + 4 * j] = v;
  }
  __syncthreads();

  // wave 0: out[b, j] = sum_k P[b][k] * (1 - 2*bit_{9-j}(k)) via 256 WMMAs
  if (wave == 0) {
    const int m = lane & 15;   // A row / B and D column
    const int half = lane >> 4;
    const float* __restrict__ base = &P[m & 7][2 * half];
    const int j = (m < 10) ? m : 0;  // B column -> qubit (cols 10..15 unused)
    const int shift = 9 - j;
    v8f acc0 = {}, acc1 = {}, acc2 = {}, acc3 = {};
#pragma unroll 2
    for (int kc = 0; kc < 256; kc += 4) {
      acc0 = gemm_step(base, shift, half, kc + 0, acc0);
      acc1 = gemm_step(base, shift, half, kc + 1, acc1);
      acc2 = gemm_step(base, shift, half, kc + 2, acc2);
      acc3 = gemm_step(base, shift, half, kc + 3, acc3);
    }
    const v8f acc = (acc0 + acc1) + (acc2 + acc3);
    // D VGPR i, lanes 0-15 = row M=i = batch i, col N = lane
    if (lane < 10) {
#pragma unroll
      for (int i = 0; i < 8; ++i)
        out[(blockIdx.x * 8 + i) * 10 + lane] = acc[i];
    }
  }
}

extern "C" void kernel_launch(void* const* d_in, const int* in_sizes, int n_in,
                              void* d_out, int out_size, void* d_ws,
                              size_t ws_size, hipStream_t stream) {
  const float* inp = (const float*)d_in[0];  // (8192, 10) f32
  const float* wts = (const float*)d_in[1];  // (2, 10)    f32
  float* out = (float*)d_out;                // (8192, 10) f32
  const int B = in_sizes[0] / 10;            // 8192
  const int blocks = B / 8;                  // 8 batch elems (waves) per block
  qsim_kernel<<<blocks, 256, 0, stream>>>(inp, wts, out);
}